// FullyConnectedGATSelfAttention_6665789243986
// MI455X (gfx1250) — compile-verified
//
#include <hip/hip_runtime.h>
#include <math.h>

#define B_DIM 64
#define E_DIM 512
#define IN_DIMS 1024
#define OUT_DIMS 256
#define NEG_INF_F (-1e30f)
#define SLOPE 0.2f

typedef __attribute__((ext_vector_type(2))) float v2f;
typedef __attribute__((ext_vector_type(8))) float v8f;

static __device__ __forceinline__ v8f wmma_f32_k4(v2f a, v2f b, v8f c) {
  // D = A(16x4) * B(4x16) + C(16x16), fp32, wave32
  return __builtin_amdgcn_wmma_f32_16x16x4_f32(
      /*neg_a=*/false, a, /*neg_b=*/false, b,
      /*c_mod=*/(short)0, c, /*reuse_a=*/false, /*reuse_b=*/false);
}

static __device__ __forceinline__ float wave_reduce_sum(float v) {
#pragma unroll
  for (int off = 16; off > 0; off >>= 1) v += __shfl_xor(v, off, 32);
  return v;
}
static __device__ __forceinline__ float wave_reduce_max(float v) {
#pragma unroll
  for (int off = 16; off > 0; off >>= 1) v = fmaxf(v, __shfl_xor(v, off, 32));
  return v;
}

// -------------------- Kernel 1: h = X @ W^T + bias (WMMA fp32) -------------
// grid: (OUT_DIMS/64, B*E/16), block: 128 (4 waves). Each block: 16 M-rows x
// 64 N-cols. A tile (16 x 64 K-chunk) staged in LDS [k][m] (conflict-free
// fragment reads), shared by the 4 waves; each wave owns one 16-wide N tile.
__global__ __launch_bounds__(128) void gat_linear_wmma(
    const float* __restrict__ X, const float* __restrict__ W,
    const float* __restrict__ Wb, float* __restrict__ h) {
  __shared__ float sA[64 * 16];

  const int tid  = threadIdx.x;
  const int lane = tid & 31;
  const int wv   = tid >> 5;                 // wave 0..3
  const int m0   = blockIdx.y * 16;
  const int n0   = blockIdx.x * 64 + wv * 16;

  const int rl   = lane & 15;                // fragment row/col selector
  const int koff = (lane >> 4) * 2;          // K sub-offset per half-wave

  const int lm = tid >> 3;                   // 0..15 cooperative-load row
  const int kq = (tid & 7) * 8;              // 0..56 cooperative-load col base

  const float* Xrow = X + (size_t)(m0 + lm) * IN_DIMS;
  const float* Wn   = W + (size_t)(n0 + rl) * IN_DIMS;

  v8f acc = {};
  for (int kc = 0; kc < IN_DIMS; kc += 64) {
    __syncthreads();
    // coalesced: each 8-thread group reads 256B of one X row
    float4 f0 = *(const float4*)(Xrow + kc + kq);
    float4 f1 = *(const float4*)(Xrow + kc + kq + 4);
    sA[(kq + 0) * 16 + lm] = f0.x;
    sA[(kq + 1) * 16 + lm] = f0.y;
    sA[(kq + 2) * 16 + lm] = f0.z;
    sA[(kq + 3) * 16 + lm] = f0.w;
    sA[(kq + 4) * 16 + lm] = f1.x;
    sA[(kq + 5) * 16 + lm] = f1.y;
    sA[(kq + 6) * 16 + lm] = f1.z;
    sA[(kq + 7) * 16 + lm] = f1.w;
    __syncthreads();
#pragma unroll
    for (int k = 0; k < 64; k += 4) {
      v2f a, bf;
      a.x  = sA[(k + koff) * 16 + rl];       // A[m=rl][k+koff]
      a.y  = sA[(k + koff + 1) * 16 + rl];
      bf.x = Wn[kc + k + koff];              // B[k][n=rl] = W[n][k]
      bf.y = Wn[kc + k + koff + 1];
      acc = wmma_f32_k4(a, bf, acc);
    }
  }
  const float bias = Wb[n0 + rl];
  const int mb = (lane >> 4) * 8;            // D: VGPR v -> M = v (+8 upper half)
#pragma unroll
  for (int v = 0; v < 8; ++v) {
    h[(size_t)(m0 + mb + v) * OUT_DIMS + n0 + rl] = acc[v] + bias;
  }
}

// -------------------- Kernel 2: s1/s2 = h . a1 / h . a2 --------------------
// one wave per (b,e) row; 256-dim dual dot product, shuffle reduction.
__global__ __launch_bounds__(256) void gat_attn_scores(
    const float* __restrict__ h, const float* __restrict__ aw,
    float* __restrict__ s1, float* __restrict__ s2) {
  const int lane = threadIdx.x & 31;
  const int row  = blockIdx.x * 8 + (threadIdx.x >> 5);
  const float* hr = h + (size_t)row * OUT_DIMS;
  float d1 = 0.f, d2 = 0.f;
#pragma unroll
  for (int t = 0; t < 8; ++t) {
    int k = lane + t * 32;
    float hv = hr[k];
    d1 += hv * aw[k];
    d2 += hv * aw[OUT_DIMS + k];
  }
  d1 = wave_reduce_sum(d1);
  d2 = wave_reduce_sum(d2);
  if (lane == 0) { s1[row] = d1; s2[row] = d2; }
}

// ---------- Kernel 3: softmax(scores) @ (h * mask)  (WMMA fp32) ------------
// grid: (E/16, B), block: 512 (16 waves). Block owns one 16-row i-tile of one
// batch. P tile (16 x 512) lives in LDS [k][i]; mask m_j folded into P,
// 1/denom folded into the D store, so the GEMM streams raw h.
__global__ __launch_bounds__(512) void gat_softmax_aggregate_wmma(
    const float* __restrict__ h, const float* __restrict__ s1,
    const float* __restrict__ s2, const int* __restrict__ node_mask,
    const float* __restrict__ a_b, float* __restrict__ out) {
  __shared__ float sP[E_DIM * 16];     // [k][i], 32 KB
  __shared__ float sMask[E_DIM];
  __shared__ float sInv[16];

  const int b    = blockIdx.y;
  const int i0   = blockIdx.x * 16;
  const int tid  = threadIdx.x;
  const int lane = tid & 31;
  const int wv   = tid >> 5;           // 0..15

  const float ab = a_b[0];

  // phase 0: masked leaky-relu scores -> sP
  {
    const int i  = tid & 15;
    const int kb = tid >> 4;           // 0..31
    const float s1v = s1[b * E_DIM + i0 + i];
    const float mi  = (node_mask[b * E_DIM + i0 + i] != 0) ? 1.f : 0.f;
#pragma unroll
    for (int p = 0; p < 16; ++p) {
      int k = kb + p * 32;
      float mj = (node_mask[b * E_DIM + k] != 0) ? 1.f : 0.f;
      if (i == 0) sMask[k] = mj;
      float sc = s1v + s2[b * E_DIM + k] + ab;
      sc = (sc >= 0.f) ? sc : SLOPE * sc;
      sc = (mi * mj > 0.f) ? sc : NEG_INF_F;
      sP[k * 16 + i] = sc;
    }
  }
  __syncthreads();

  // phase 1: wave wv owns row i=wv: max, exponentiate in place (* m_j), denom
  {
    const int i = wv;
    float vals[16];
    float mx = -INFINITY;
#pragma unroll
    for (int t = 0; t < 16; ++t) {
      int k = lane + t * 32;
      vals[t] = sP[k * 16 + i];
      mx = fmaxf(mx, vals[t]);
    }
    mx = wave_reduce_max(mx);
    float sum = 0.f;
#pragma unroll
    for (int t = 0; t < 16; ++t) {
      int k = lane + t * 32;
      float e = expf(vals[t] - mx);    // masked i: all-equal -> uniform, as ref
      sum += e;
      sP[k * 16 + i] = e * sMask[k];   // fold m_j into P
    }
    sum = wave_reduce_sum(sum);
    if (lane == 0) sInv[i] = 1.f / sum;
  }
  __syncthreads();

  // phase 2: out[i-tile, n-tile(wv)] = (P @ h_b) * inv_denom
  const int rl   = lane & 15;
  const int koff = (lane >> 4) * 2;
  const int n0   = wv * 16;
  const float* hb = h + (size_t)b * E_DIM * OUT_DIMS;
  v8f acc = {};
#pragma unroll 4
  for (int k0 = 0; k0 < E_DIM; k0 += 4) {
    v2f a, bf;
    a.x  = sP[(k0 + koff) * 16 + rl];
    a.y  = sP[(k0 + koff + 1) * 16 + rl];
    bf.x = hb[(size_t)(k0 + koff) * OUT_DIMS + n0 + rl];     // 64B coalesced
    bf.y = hb[(size_t)(k0 + koff + 1) * OUT_DIMS + n0 + rl];
    acc = wmma_f32_k4(a, bf, acc);
  }
  const int mb = (lane >> 4) * 8;
#pragma unroll
  for (int v = 0; v < 8; ++v) {
    int i = mb + v;
    out[((size_t)b * E_DIM + i0 + i) * OUT_DIMS + n0 + rl] = acc[v] * sInv[i];
  }
}

// --------------------------------- launch ----------------------------------
extern "C" void kernel_launch(void* const* d_in, const int* in_sizes, int n_in,
                              void* d_out, int out_size, void* d_ws, size_t ws_size,
                              hipStream_t stream) {
  const float* X  = (const float*)d_in[0];
  // d_in[1] = adj: all-ones, unused by the reference computation
  const int*   nm = (const int*)d_in[2];
  const float* Ww = (const float*)d_in[3];
  const float* Wb = (const float*)d_in[4];
  const float* aw = (const float*)d_in[5];
  const float* ab = (const float*)d_in[6];
  float* out = (float*)d_out;

  float* h  = (float*)d_ws;                              // B*E*OUT  (32 MB)
  float* s1 = h  + (size_t)B_DIM * E_DIM * OUT_DIMS;     // B*E
  float* s2 = s1 + (size_t)B_DIM * E_DIM;                // B*E

  dim3 g1(OUT_DIMS / 64, (B_DIM * E_DIM) / 16);
  gat_linear_wmma<<<g1, 128, 0, stream>>>(X, Ww, Wb, h);

  gat_attn_scores<<<(B_DIM * E_DIM) / 8, 256, 0, stream>>>(h, aw, s1, s2);

  dim3 g3(E_DIM / 16, B_DIM);
  gat_softmax_aggregate_wmma<<<g3, 512, 0, stream>>>(h, s1, s2, nm, ab, out);
}